// DDSPDecoder_1125281432037
// MI455X (gfx1250) — compile-verified
//
#include <hip/hip_runtime.h>
#include <hip/hip_bf16.h>

#define SRATE   16000
#define TFRAMES 1000
#define NSAMP   64000
#define NHARM   64
#define NBAND   65
#define RLEN    48000
#define BATCH   8
#define NYQF    8000.0f
#define TWO_PI_SR 3.9269908169872414e-4f   /* 2*pi/16000 */
#define TWO_PI_128 4.9087385212340526e-2f  /* 2*pi/128  */
#define LOG2_440 8.7813597135246599f
#define LN10 2.3025850929940457f

#define RPAD    16
#define RBUFN   48272                      /* 48000 + 16 front + chunk spill */
#define APAD    256
#define APITCH  (NSAMP + APAD)             /* 64256 per batch */
#define MC      16                         /* m-blocks per LDS chunk */
#define CH      272                        /* 16*MC + 16 staged floats  */

typedef __attribute__((ext_vector_type(2))) float v2f;
typedef __attribute__((ext_vector_type(8))) float v8f;

__device__ __forceinline__ float sigm(float x) { return 1.0f / (1.0f + expf(-x)); }
__device__ __forceinline__ float exp_sig(float x) {
  return 2.0f * powf(sigm(x), LN10) + 1e-7f;
}

// ---------------------------------------------------------------- pre-net
__global__ void prenet_kernel(const float* __restrict__ f0,
                              const float* __restrict__ loud,
                              const float* __restrict__ pw,
                              const float* __restrict__ pb,
                              float* __restrict__ xout) {
  int idx = blockIdx.x * blockDim.x + threadIdx.x;
  if (idx >= BATCH * TFRAMES * 128) return;
  int j = idx & 127;
  int bt = idx >> 7;
  float f = fminf(fmaxf(f0[bt], 1.f), 8000.f);
  float f0m = 12.f * (log2f(f) - LOG2_440) + 69.f;
  f0m = fminf(fmaxf(f0m, 24.f), 84.f);
  float v = f0m * pw[j] + loud[bt] * pw[128 + j] + pb[j];
  xout[idx] = fmaxf(v, 0.f);
}

// ------------------------------------------------ generic WMMA f32 GEMM
// C[M,N] = act(A[M,K] @ W[K,N] + bias[N]);   one wave32 per 16x16 tile.
// A 16x4 layout: lanes0-15 hold K={0,1}, lanes16-31 K={2,3} (ISA 7.12.2).
// B 4x16 / C 16x16: row-striped across lanes; VGPR i = rows i, i + rows/2.
__global__ __launch_bounds__(32)
void gemm_wmma_kernel(const float* __restrict__ A, const float* __restrict__ W,
                      const float* __restrict__ bias, float* __restrict__ C,
                      int M, int N, int K, int relu) {
  int nT = N >> 4;
  int tn = blockIdx.x % nT;
  int tm = blockIdx.x / nT;
  int lane = threadIdx.x;
  int half = lane >> 4;
  int lm = lane & 15;
  int row = tm * 16 + lm;
  int col = tn * 16 + lm;
  v8f c = {};
  for (int k0 = 0; k0 < K; k0 += 4) {
    int ka = k0 + 2 * half;
    float a0 = A[(size_t)row * K + ka];
    float a1 = A[(size_t)row * K + ka + 1];
    float b0 = W[(size_t)ka * N + col];
    float b1 = W[(size_t)(ka + 1) * N + col];
    v2f av = {a0, a1};
    v2f bv = {b0, b1};
    c = __builtin_amdgcn_wmma_f32_16x16x4_f32(false, av, false, bv,
                                              (short)0, c, false, false);
  }
  float bc = bias[col];
#pragma unroll
  for (int i = 0; i < 8; ++i) {
    int r = tm * 16 + i + 8 * half;
    float v = c[i] + bc;
    if (relu) v = fmaxf(v, 0.f);
    C[(size_t)r * N + col] = v;
  }
}

// ---------------------------------------------------------------- GRU scan
__global__ __launch_bounds__(256)
void gru_kernel(const float* __restrict__ xp, const float* __restrict__ rk,
                const float* __restrict__ grub, float* __restrict__ hseq) {
  __shared__ float h[256];
  const int b = blockIdx.x;
  const int j = threadIdx.x;
  h[j] = 0.f;
  __syncthreads();
  const float bz = grub[768 + j];
  const float br = grub[768 + 256 + j];
  const float bh = grub[768 + 512 + j];
  for (int t = 0; t < TFRAMES; ++t) {
    float rz = bz, rr = br, rh = bh;
    for (int i = 0; i < 256; ++i) {
      float hv = h[i];
      rz = fmaf(hv, rk[i * 768 + j], rz);
      rr = fmaf(hv, rk[i * 768 + 256 + j], rr);
      rh = fmaf(hv, rk[i * 768 + 512 + j], rh);
    }
    const float* xrow = xp + ((size_t)b * TFRAMES + t) * 768;
    float z = sigm(xrow[j] + rz);
    float r = sigm(xrow[256 + j] + rr);
    float hh = tanhf(xrow[512 + j] + r * rh);
    float hn = z * h[j] + (1.f - z) * hh;
    __syncthreads();
    h[j] = hn;
    hseq[((size_t)b * TFRAMES + t) * 256 + j] = hn;
    __syncthreads();
  }
}

// ----------------------- heads: amp / harm softmax / noise mags -> time IR
__global__ __launch_bounds__(256)
void heads_kernel(const float* __restrict__ y2, const float* __restrict__ f0,
                  const float* __restrict__ amp_w, const float* __restrict__ amp_b,
                  const float* __restrict__ harm_w, const float* __restrict__ harm_b,
                  const float* __restrict__ noise_w, const float* __restrict__ noise_b,
                  float* __restrict__ amp_hd, float* __restrict__ irn) {
  __shared__ float xr[128];
  __shared__ float hl[64];
  __shared__ float mg[65];
  __shared__ float hd[64];
  __shared__ float smax, ssum, scale;
  int bt = blockIdx.x;
  int tid = threadIdx.x;
  if (tid < 128) xr[tid] = y2[(size_t)bt * 128 + tid];
  __syncthreads();
  if (tid < 64) {
    float acc = harm_b[tid];
    for (int i = 0; i < 128; ++i) acc = fmaf(xr[i], harm_w[i * 64 + tid], acc);
    hl[tid] = acc;
  } else if (tid < 129) {
    int cidx = tid - 64;
    float acc = noise_b[cidx];
    for (int i = 0; i < 128; ++i) acc = fmaf(xr[i], noise_w[i * 65 + cidx], acc);
    mg[cidx] = exp_sig(exp_sig(acc) - 5.f);   // head scale_fn then -5 bias scale_fn
  } else if (tid == 129) {
    float acc = amp_b[0];
    for (int i = 0; i < 128; ++i) acc = fmaf(xr[i], amp_w[i], acc);
    scale = exp_sig(exp_sig(acc));            // amp (double exp_sigmoid)
  }
  __syncthreads();
  if (tid == 0) {
    float m = hl[0];
    for (int i = 1; i < 64; ++i) m = fmaxf(m, hl[i]);
    float s = 0.f;
    for (int i = 0; i < 64; ++i) s += expf(hl[i] - m);
    smax = m; ssum = s;
  }
  __syncthreads();
  float f0v = f0[bt];
  if (tid < 64) {
    float sm = expf(hl[tid] - smax) / ssum;
    float v = exp_sig(sm);
    if (f0v * (float)(tid + 1) >= NYQF) v = 0.f;  // remove_above_nyquist
    hd[tid] = v;
  }
  __syncthreads();
  if (tid == 0) {
    float d = 0.f;
    for (int i = 0; i < 64; ++i) d += hd[i];
    if (d == 0.f) d = 1e-7f;
    scale = scale / d;                         // amp / safe_den
  }
  __syncthreads();
  if (tid < 64) amp_hd[(size_t)bt * 64 + tid] = hd[tid] * scale;
  if (tid < 128) {
    // ir = fftshift(irfft(mags,128) * fftshift(hann)),  final[n] = ir0[(n+64)%128]*hann[n]
    int n = tid;
    int m_ = (n + 64) & 127;
    float acc = mg[0] + ((m_ & 1) ? -mg[64] : mg[64]);
    for (int k = 1; k < 64; ++k)
      acc += 2.f * mg[k] * cosf(TWO_PI_128 * (float)(k * m_));
    float w = 0.5f - 0.5f * cosf(TWO_PI_128 * (float)n);
    irn[(size_t)bt * 128 + n] = (acc * (1.f / 128.f)) * w;
  }
}

// ------------------- harmonic phase: closed-form per-frame segment sums
__global__ void segsum_kernel(const float* __restrict__ f0, float* __restrict__ seg) {
  int idx = blockIdx.x * blockDim.x + threadIdx.x;
  if (idx >= BATCH * NHARM * TFRAMES) return;
  int t = idx % TFRAMES;
  int bh = idx / TFRAMES;
  int h = bh & 63;
  int b = bh >> 6;
  float f0t = f0[b * TFRAMES + t];
  int tn = (t < TFRAMES - 1) ? t + 1 : TFRAMES - 1;
  float f0n = f0[b * TFRAMES + tn];
  // sum_{q=0..63} (f0t + (f0n-f0t)*q/64) = 64*f0t + 31.5*(f0n-f0t)
  float s = 64.f * f0t + 31.5f * (f0n - f0t);
  seg[idx] = TWO_PI_SR * (float)(h + 1) * s;
}

__global__ void prefix_kernel(const float* __restrict__ seg, float* __restrict__ pre) {
  int row = blockIdx.x * blockDim.x + threadIdx.x;
  if (row >= BATCH * NHARM) return;
  const float* s = seg + (size_t)row * TFRAMES;
  float* p = pre + (size_t)row * TFRAMES;
  float run = 0.f;
  for (int t = 0; t < TFRAMES; ++t) { p[t] = run; run += s[t]; }
}

// ---------------------------------------------------------- harmonic synth
__global__ __launch_bounds__(64)
void hsynth_kernel(const float* __restrict__ f0, const float* __restrict__ amp_hd,
                   const float* __restrict__ pre, float* __restrict__ audio_h) {
  __shared__ float a0[64], a1[64], sp[64];
  int bt = blockIdx.x;
  int b = bt / TFRAMES;
  int t = bt % TFRAMES;
  int tn = (t < TFRAMES - 1) ? t + 1 : TFRAMES - 1;
  int q = threadIdx.x;
  a0[q] = amp_hd[(size_t)bt * 64 + q];
  a1[q] = amp_hd[((size_t)b * TFRAMES + tn) * 64 + q];
  sp[q] = pre[((size_t)(b * 64 + q)) * TFRAMES + t];
  __syncthreads();
  float f0t = f0[b * TFRAMES + t];
  float f0n = f0[b * TFRAMES + tn];
  float frac = (float)q * (1.f / 64.f);
  float f0i = f0t + (f0n - f0t) * frac;
  // inclusive partial sum inside frame: (q+1)*f0t + (f0n-f0t)*q(q+1)/128
  float cc = (float)(q + 1) * f0t + (f0n - f0t) * ((float)(q * (q + 1)) * (1.f / 128.f));
  float acc = 0.f;
  for (int h = 0; h < 64; ++h) {
    float fr = f0i * (float)(h + 1);
    if (fr < NYQF) {
      float a = a0[h] + (a1[h] - a0[h]) * frac;
      float ph = sp[h] + TWO_PI_SR * (float)(h + 1) * cc;
      acc = fmaf(a, sinf(ph), acc);
    }
  }
  audio_h[(size_t)b * NSAMP + t * 64 + q] = acc;
}

// --------------------------------------- filtered noise (128-tap time conv)
__global__ void nconv_kernel(const float* __restrict__ noise,
                             const float* __restrict__ irn,
                             float* __restrict__ audio_n) {
  int idx = blockIdx.x * blockDim.x + threadIdx.x;
  if (idx >= BATCH * NSAMP) return;
  int b = idx / NSAMP;
  int s = idx % NSAMP;
  float acc = 0.f;
  for (int k = 0; k < 128; ++k) {
    int u = s + 62 - k;                       // delay comp start = 62
    if ((unsigned)u < (unsigned)NSAMP)
      acc = fmaf(noise[(size_t)b * NSAMP + u],
                 irn[((size_t)b * TFRAMES + (u >> 6)) * 128 + k], acc);
  }
  audio_n[idx] = acc;
}

// --------------------------- padded reverb IR:  rbuf[i+16] = masked rir[i]
__global__ void rpad_kernel(const float* __restrict__ rir, float* __restrict__ rbuf) {
  int idx = blockIdx.x * blockDim.x + threadIdx.x;
  if (idx >= RBUFN) return;
  int i = idx - RPAD;
  rbuf[idx] = (i >= 1 && i < RLEN) ? rir[i] : 0.f;   // i>=1 zeroes rir[0] (dry mask)
}

// --------- mix: dry output + zero-padded audio buffer for the reverb GEMM
__global__ void mix_kernel(const float* __restrict__ ah, const float* __restrict__ an,
                           float* __restrict__ amixp, float* __restrict__ outp) {
  int idx = blockIdx.x * blockDim.x + threadIdx.x;
  if (idx >= BATCH * NSAMP) return;
  int b = idx / NSAMP;
  int s = idx % NSAMP;
  float v = ah[idx] + an[idx];
  amixp[(size_t)b * APITCH + APAD + s] = v;
  outp[idx] = v;                               // dry path of add_dry reverb
  if (idx < BATCH * APAD)                      // zero the front pads
    amixp[(size_t)(idx / APAD) * APITCH + (idx % APAD)] = 0.f;
}

// ----------------------------- reverb as Toeplitz block-GEMM via WMMA f32
// wet[b,16p+q] = sum_m sum_{j=0..15} r[16m+q-j] * a[b,16(p-m)+j]
// M=q(16), K=j(16 -> 4 chained 16x16x4), N=16 = 8 batches x 2 p-tiles.
// Operands staged in LDS per chunk of MC m-blocks (guard-free via padding).
__global__ __launch_bounds__(32)
void reverb_wmma_kernel(const float* __restrict__ rbuf,
                        const float* __restrict__ abuf,
                        float* __restrict__ outp) {
  __shared__ float lr[CH];
  __shared__ float la[BATCH * CH];
  int tile = blockIdx.x >> 3;                  // 2000 p-pair tiles
  int split = blockIdx.x & 7;                  // split-K over m
  int P = tile * 2;
  int lane = threadIdx.x;
  int half = lane >> 4;
  int lm = lane & 15;
  int bN = lm & 7;                             // batch in N lanes
  int dp = lm >> 3;                            // p-tile select in N lanes
  int pn = P + dp;
  int mStart = split * 376;                    // 3001 m-blocks total (k spill)
  int mEnd = mStart + 376;
  if (mEnd > 3001) mEnd = 3001;
  if (mEnd > P + 2) mEnd = P + 2;              // causal: m <= p+1
  if (mStart >= mEnd) return;
  v8f c = {};
  for (int m0 = mStart; m0 < mEnd; m0 += MC) {
    int rb0 = 16 * m0;                                   // rbuf chunk base
    int ab0 = 16 * (P - m0 - (MC - 1)) + APAD;           // abuf chunk base
    __syncthreads();
    for (int i = lane; i < CH; i += 32) lr[i] = rbuf[rb0 + i];
    __builtin_prefetch(rbuf + rb0 + CH, 0, 0);
    for (int i = lane; i < BATCH * CH; i += 32)
      la[i] = abuf[(size_t)(i / CH) * APITCH + ab0 + (i % CH)];
    __syncthreads();
    int mlim = mEnd - m0;
    if (mlim > MC) mlim = MC;
    for (int dm = 0; dm < mlim; ++dm) {
      int ra = 16 * dm + lm + RPAD;                      // lr index (j = 0)
      int ao = bN * CH + 16 * (dp - dm + (MC - 1));      // la index (j = 0)
#pragma unroll
      for (int jj = 0; jj < 4; ++jj) {
        int j = 4 * jj + 2 * half;                       // K-lane split (ISA)
        float a0 = lr[ra - j];
        float a1 = lr[ra - j - 1];
        float b0 = la[ao + j];
        float b1 = la[ao + j + 1];
        v2f av = {a0, a1};
        v2f bv = {b0, b1};
        c = __builtin_amdgcn_wmma_f32_16x16x4_f32(false, av, false, bv,
                                                  (short)0, c, false, false);
      }
    }
  }
#pragma unroll
  for (int i = 0; i < 8; ++i) {
    int s = 16 * pn + i + 8 * half;
    atomicAdd(&outp[(size_t)bN * NSAMP + s], c[i]);
  }
}

// -------------------------------------------------------------------------
extern "C" void kernel_launch(void* const* d_in, const int* in_sizes, int n_in,
                              void* d_out, int out_size, void* d_ws, size_t ws_size,
                              hipStream_t stream) {
  (void)in_sizes; (void)n_in; (void)out_size; (void)ws_size;
  const float* f0       = (const float*)d_in[0];
  const float* loud     = (const float*)d_in[1];
  const float* noise    = (const float*)d_in[2];
  const float* pre_w    = (const float*)d_in[3];
  const float* pre_b    = (const float*)d_in[4];
  const float* gru_k    = (const float*)d_in[5];
  const float* gru_rk   = (const float*)d_in[6];
  const float* gru_b    = (const float*)d_in[7];
  const float* post_w1  = (const float*)d_in[8];
  const float* post_b1  = (const float*)d_in[9];
  const float* post_w2  = (const float*)d_in[10];
  const float* post_b2  = (const float*)d_in[11];
  const float* amp_w    = (const float*)d_in[12];
  const float* amp_b    = (const float*)d_in[13];
  const float* harm_w   = (const float*)d_in[14];
  const float* harm_b   = (const float*)d_in[15];
  const float* noise_w  = (const float*)d_in[16];
  const float* noise_b  = (const float*)d_in[17];
  const float* rir      = (const float*)d_in[18];
  float* outp = (float*)d_out;

  float* ws = (float*)d_ws;
  size_t o = 0;
  float* x_pre   = ws + o; o += (size_t)8000 * 128;   // 1,024,000
  float* xp      = ws + o; o += (size_t)8000 * 768;   // 6,144,000
  float* h_seq   = ws + o; o += (size_t)8000 * 256;   // 2,048,000
  float* y1      = ws + o; o += (size_t)8000 * 256;   // 2,048,000
  float* y2      = ws + o; o += (size_t)8000 * 128;   // 1,024,000
  float* amp_hd  = ws + o; o += (size_t)8000 * 64;    //   512,000
  float* irn     = ws + o; o += (size_t)8000 * 128;   // 1,024,000
  float* seg     = ws + o; o += (size_t)512 * 1000;   //   512,000
  float* segpre  = ws + o; o += (size_t)512 * 1000;   //   512,000
  float* audio_h = ws + o; o += (size_t)BATCH * NSAMP;
  float* audio_n = ws + o; o += (size_t)BATCH * NSAMP;
  float* amixp   = ws + o; o += (size_t)BATCH * APITCH;
  float* rbuf    = ws + o; o += (size_t)RBUFN;

  prenet_kernel<<<(8 * 1000 * 128 + 255) / 256, 256, 0, stream>>>(
      f0, loud, pre_w, pre_b, x_pre);

  gemm_wmma_kernel<<<500 * 48, 32, 0, stream>>>(x_pre, gru_k, gru_b, xp,
                                                8000, 768, 128, 0);
  gru_kernel<<<8, 256, 0, stream>>>(xp, gru_rk, gru_b, h_seq);

  gemm_wmma_kernel<<<500 * 16, 32, 0, stream>>>(h_seq, post_w1, post_b1, y1,
                                                8000, 256, 256, 1);
  gemm_wmma_kernel<<<500 * 8, 32, 0, stream>>>(y1, post_w2, post_b2, y2,
                                               8000, 128, 256, 1);

  heads_kernel<<<8000, 256, 0, stream>>>(y2, f0, amp_w, amp_b, harm_w, harm_b,
                                         noise_w, noise_b, amp_hd, irn);

  segsum_kernel<<<(512 * 1000 + 255) / 256, 256, 0, stream>>>(f0, seg);
  prefix_kernel<<<2, 256, 0, stream>>>(seg, segpre);
  hsynth_kernel<<<8000, 64, 0, stream>>>(f0, amp_hd, segpre, audio_h);

  nconv_kernel<<<(BATCH * NSAMP + 255) / 256, 256, 0, stream>>>(noise, irn, audio_n);

  rpad_kernel<<<(RBUFN + 255) / 256, 256, 0, stream>>>(rir, rbuf);
  mix_kernel<<<(BATCH * NSAMP + 255) / 256, 256, 0, stream>>>(audio_h, audio_n,
                                                              amixp, outp);

  reverb_wmma_kernel<<<2000 * 8, 32, 0, stream>>>(rbuf, amixp, outp);
}